// ProximityAwareLoss_25786983645935
// MI455X (gfx1250) — compile-verified
//
#include <hip/hip_runtime.h>
#include <stdint.h>

// Problem dims (fixed by reference setup_inputs)
#define B_      256
#define S_      32768
#define C_      3
#define TILE    2048            // positions per DMA tile
#define NTILES  (S_ / TILE)     // 16
#define THREADS 1024            // 32 waves (wave32)
#define NWAVES  (THREADS / 32)

typedef uint32_t v4u __attribute__((ext_vector_type(4)));
typedef int      v4i __attribute__((ext_vector_type(4)));
typedef int      v8i __attribute__((ext_vector_type(8)));

// ---------------- LDS layout (bytes) ----------------
// double-buffered logits staging: 2 * 2048 * 3 * 4 = 49152
#define LOG_OFF   0
// double-buffered labels staging: 2 * 2048 * 4 = 16384
#define LBL_OFF   (LOG_OFF + 2 * TILE * C_ * 4)
// whole-row CE values: 32768 * 4 = 131072
#define CE_OFF    (LBL_OFF + 2 * TILE * 4)
// bit words (S_/32 = 1024 words + 2 zero pads), 3 arrays
#define SW_OFF    (CE_OFF + S_ * 4)
#define PR_OFF    (SW_OFF + 4112)
#define VA_OFF    (PR_OFF + 4112)
// small tables + reduction scratch
#define DW_OFF    (VA_OFF + 4112)       // 11 floats
#define CW_OFF    (DW_OFF + 48)         // 3 floats
#define RED_OFF   (CW_OFF + 16)         // 96 floats + 32 uints
#define SMEM_BYTES (RED_OFF + 512)      // ~205 KB, fits 320 KB WGP LDS

// ---------------- TDM: 1-D DMA tile (global -> LDS) ----------------
// D# group0: count=1, lds_addr, global_addr[56:0], type=2 ("image")
// D# group1: data_size=4B, tensor_dim0 = tile_dim0 = nelem, tensor_dim1 = tile_dim1 = 1
__device__ __forceinline__ void tdm_load_1d(uint32_t lds_byte_off,
                                            const void* gptr,
                                            uint32_t nelem_dwords) {
  uint64_t ga = (uint64_t)(uintptr_t)gptr;
  v4u g0;
  g0.x = 1u;                                                  // count=1, user mode
  g0.y = lds_byte_off;                                        // lds_addr (bytes)
  g0.z = (uint32_t)ga;                                        // global_addr[31:0]
  g0.w = ((uint32_t)(ga >> 32) & 0x01FFFFFFu) | (2u << 30);   // addr[56:32] | type=2
  v8i g1;
  g1[0] = (int)(2u << 16);                                    // data_size = 4B
  g1[1] = (int)((nelem_dwords & 0xFFFFu) << 16);              // tensor_dim0[15:0]
  g1[2] = (int)(((nelem_dwords >> 16) & 0xFFFFu) | (1u << 16)); // td0 hi | tensor_dim1 lo = 1
  g1[3] = (int)((nelem_dwords & 0xFFFFu) << 16);              // tile_dim0 (<= 65535)
  g1[4] = 1;                                                  // tile_dim1 = 1
  g1[5] = (int)nelem_dwords;                                  // tensor_dim0_stride lo32
  g1[6] = 0;
  g1[7] = 0;
  v4i z = {0, 0, 0, 0};
#if __clang_major__ >= 23
  v8i z8 = {0, 0, 0, 0, 0, 0, 0, 0};
  __builtin_amdgcn_tensor_load_to_lds(g0, g1, z, z, z8, 0);
#else
  __builtin_amdgcn_tensor_load_to_lds(g0, g1, z, z, 0);
#endif
}

// ---------------- wave32 reductions ----------------
__device__ __forceinline__ float wave_sum(float v) {
  for (int o = 16; o > 0; o >>= 1) v += __shfl_down(v, o, 32);
  return v;
}
__device__ __forceinline__ uint32_t wave_or(uint32_t v) {
  for (int o = 16; o > 0; o >>= 1) v |= (uint32_t)__shfl_down((int)v, o, 32);
  return v;
}
__device__ __forceinline__ int imin_(int a, int b) { return a < b ? a : b; }

// ---------------- kernel 1: one workgroup per row ----------------
__global__ __launch_bounds__(THREADS)
void prox_loss_rows(const float* __restrict__ logits,
                    const int*   __restrict__ labels,
                    const float* __restrict__ dweights,
                    const float* __restrict__ cweights,
                    float* __restrict__ row_sum,
                    float* __restrict__ row_cnt)
{
  extern __shared__ char smem[];
  float*    LOG = (float*)(smem + LOG_OFF);
  int*      LBL = (int*)(smem + LBL_OFF);
  float*    CE  = (float*)(smem + CE_OFF);
  uint32_t* SW  = (uint32_t*)(smem + SW_OFF);   // [0]=pad, [1..1024], [1025]=pad
  uint32_t* PR  = (uint32_t*)(smem + PR_OFF);
  uint32_t* VA  = (uint32_t*)(smem + VA_OFF);
  float*    DW  = (float*)(smem + DW_OFF);
  float*    CW  = (float*)(smem + CW_OFF);
  float*    RED = (float*)(smem + RED_OFF);               // 96 floats
  uint32_t* REDU = (uint32_t*)(smem + RED_OFF + 96 * 4);  // 32 uints

  const int t    = threadIdx.x;
  const int row  = blockIdx.x;
  const int lane = t & 31;
  const int wid  = t >> 5;

  if (t < 11) DW[t] = dweights[t];
  if (t < 3)  CW[t] = cweights[t];
  if (t == 0) {
    SW[0] = 0; SW[S_ / 32 + 1] = 0;
    PR[0] = 0; PR[S_ / 32 + 1] = 0;
    VA[0] = 0; VA[S_ / 32 + 1] = 0;
  }

  const float* grow = logits + (size_t)row * S_ * C_;
  const int*   lrow = labels + (size_t)row * S_;

  // prologue: wave 0 kicks off DMA for tile 0 (2 descriptors in flight)
  if (wid == 0) {
    tdm_load_1d(LOG_OFF, grow, TILE * C_);
    tdm_load_1d(LBL_OFF, lrow, TILE);
  }

  // ---- phase 1: stream tiles through LDS, compute CE + ballot bit words ----
  for (int j = 0; j < NTILES; ++j) {
    const int buf = j & 1;
    if (wid == 0) {
      if (j + 1 < NTILES) {
        const int nb = (j + 1) & 1;
        tdm_load_1d(LOG_OFF + nb * TILE * C_ * 4, grow + (size_t)(j + 1) * TILE * C_, TILE * C_);
        tdm_load_1d(LBL_OFF + nb * TILE * 4,      lrow + (j + 1) * TILE,              TILE);
        __builtin_amdgcn_s_wait_tensorcnt(2);   // tile j's two DMAs are complete
      } else {
        __builtin_amdgcn_s_wait_tensorcnt(0);
      }
    }
    __syncthreads();  // publish DMA'd LDS to all waves

    const float* lg = LOG + buf * TILE * C_;
    const int*   lb = LBL + buf * TILE;
    const int P = j * TILE;
    #pragma unroll
    for (int g = 0; g < 2; ++g) {
      const int q   = g * 1024 + t;
      const int lbl = lb[q];
      const float l0 = lg[3 * q + 0];
      const float l1 = lg[3 * q + 1];
      const float l2 = lg[3 * q + 2];
      const bool valid = (lbl != -100);
      const int  safe  = valid ? lbl : 0;
      const float m   = fmaxf(l0, fmaxf(l1, l2));
      const float lse = m + logf(expf(l0 - m) + expf(l1 - m) + expf(l2 - m));
      const float pick = (safe == 0) ? l0 : ((safe == 1) ? l1 : l2);
      const float ce0  = valid ? (lse - pick) * CW[safe] : 0.0f;
      int pred = 0; float best = l0;
      if (l1 > best) { best = l1; pred = 1; }   // strict > == argmax first-max tie rule
      if (l2 > best) { best = l2; pred = 2; }
      const bool swb = (lbl == 1) || (lbl == 2);
      const bool prb = (pred != 0);
      CE[P + q] = ce0;
      const uint32_t bs = (uint32_t)__ballot(swb);
      const uint32_t bp = (uint32_t)__ballot(prb);
      const uint32_t bv = (uint32_t)__ballot(valid);
      if (lane == 0) {
        const int wirow = (P + q) >> 5;  // wave-uniform
        SW[1 + wirow] = bs;
        PR[1 + wirow] = bp;
        VA[1 + wirow] = bv;
      }
    }
    __syncthreads();  // reads of buf done before wave0 re-targets it next iter
  }

  // ---- phase 2: +/-10 window distance logic + accumulation ----
  float accB = 0.f, accE = 0.f, accC = 0.f;
  uint32_t orp = 0;
  #pragma unroll 4
  for (int k = 0; k < 32; ++k) {
    const int p  = k * 1024 + t;        // coalesced over lanes
    const int wi = p >> 5;              // wave-uniform word index
    const float ce0 = CE[p];
    const uint32_t swl = SW[wi], swc = SW[1 + wi], swh = SW[2 + wi];
    const uint32_t prl = PR[wi], prc = PR[1 + wi], prh = PR[2 + wi];
    const uint32_t vac = VA[1 + wi];
    orp |= prc;
    // 52-bit windows: bit (10 + i) of W == bit i of current word
    const uint64_t Wsw = ((uint64_t)(swl >> 22)) | ((uint64_t)swc << 10) | ((uint64_t)swh << 42);
    const uint64_t Wpr = ((uint64_t)(prl >> 22)) | ((uint64_t)prc << 10) | ((uint64_t)prh << 42);
    const uint32_t vt = (uint32_t)(Wsw >> lane) & 0x1FFFFFu;  // positions p-10..p+10
    const uint32_t vp = (uint32_t)(Wpr >> lane) & 0x1FFFFFu;
    const bool swb = (swc >> lane) & 1u;
    const bool prb = (prc >> lane) & 1u;
    const bool vab = (vac >> lane) & 1u;

    float bonus = 0.f;
    if (prb && vt) {                    // pred-switch near a true switch (d<=10)
      const uint32_t below = vt & 0x7FFu;  // p-10..p  (bit 10 == p)
      const uint32_t above = vt >> 10;     // p..p+10  (bit 0  == p)
      const int db = below ? (__builtin_clz(below) - 21) : 1000; // 10 - msb
      const int da = above ? __builtin_ctz(above) : 1000;
      bonus = ce0 * DW[imin_(db, da)] * 0.5f;  // ce0 already 0 when invalid
    }
    if (swb && vp == 0u) accE += ce0;   // true switch with no pred within 10
    const float fp = (prb && !swb && vab) ? 2.0f : 0.0f;  // FP_PENALTY
    accB += ce0 - bonus + fp;
    accC += vab ? 1.0f : 0.0f;
  }

  accB = wave_sum(accB);
  accE = wave_sum(accE);
  accC = wave_sum(accC);
  orp  = wave_or(orp);
  if (lane == 0) { RED[wid] = accB; RED[32 + wid] = accE; RED[64 + wid] = accC; REDU[wid] = orp; }
  __syncthreads();
  if (t == 0) {
    float sb = 0.f, se = 0.f, sc = 0.f; uint32_t so = 0u;
    for (int i = 0; i < NWAVES; ++i) { sb += RED[i]; se += RED[32 + i]; sc += RED[64 + i]; so |= REDU[i]; }
    row_sum[row] = sb + (so ? se : 0.0f);   // has_pred gates the doubling term
    row_cnt[row] = sc;
  }
}

// ---------------- kernel 2: reduce 256 rows, divide ----------------
__global__ __launch_bounds__(B_)
void prox_loss_final(const float* __restrict__ row_sum,
                     const float* __restrict__ row_cnt,
                     float* __restrict__ out)
{
  __shared__ float sb[B_], sc[B_];
  const int t = threadIdx.x;
  sb[t] = row_sum[t];
  sc[t] = row_cnt[t];
  __syncthreads();
  for (int s = B_ / 2; s > 0; s >>= 1) {
    if (t < s) { sb[t] += sb[t + s]; sc[t] += sc[t + s]; }
    __syncthreads();
  }
  if (t == 0) out[0] = sb[0] / fmaxf(sc[0], 1.0f);
}

extern "C" void kernel_launch(void* const* d_in, const int* in_sizes, int n_in,
                              void* d_out, int out_size, void* d_ws, size_t ws_size,
                              hipStream_t stream) {
  (void)in_sizes; (void)n_in; (void)out_size; (void)ws_size;
  const float* logits = (const float*)d_in[0];
  const int*   labels = (const int*)d_in[1];
  const float* dw     = (const float*)d_in[2];
  const float* cw     = (const float*)d_in[3];
  float* wsf     = (float*)d_ws;
  float* row_sum = wsf;
  float* row_cnt = wsf + B_;
  prox_loss_rows<<<B_, THREADS, SMEM_BYTES, stream>>>(logits, labels, dw, cw, row_sum, row_cnt);
  prox_loss_final<<<1, B_, 0, stream>>>(row_sum, row_cnt, (float*)d_out);
}